// Qwen2DecoderLayer_11725260718471
// MI455X (gfx1250) — compile-verified
//
#include <hip/hip_runtime.h>
#include <math.h>
#include <stdint.h>

// ---------------- config (Qwen2-1.5B decode) ----------------
#define HID   1536
#define NH    12
#define NKV   2
#define HD    128
#define INTER 8960
#define B_    64
#define MAXB  256
#define BS_   16
#define NBLK  (B_ * MAXB)
#define LMAX  (MAXB * BS_)          // 4096
#define Q_SZ  (NH * HD)             // 1536
#define KV_SZ (NKV * HD)            // 256
#define QKV_N (Q_SZ + 2 * KV_SZ)    // 2048
#define GQA_G (NH / NKV)            // 6
#define EPS_  1e-6f
#define ATT_SCALE 0.08838834764831844f   // 1/sqrt(128)
#define TK    32                    // GEMM K-tile (rows of B per async stage)

typedef __attribute__((ext_vector_type(2))) float v2f;
typedef __attribute__((ext_vector_type(8))) float v8f;

// ---------------- fused residual add + RMSNorm ----------------
__global__ void add_rmsnorm_kernel(const float* __restrict__ x,
                                   const float* __restrict__ r,
                                   const float* __restrict__ w,
                                   float* __restrict__ res_out,
                                   float* __restrict__ h_out) {
    const int b = blockIdx.x;
    const int t = threadIdx.x;            // 256 threads, HID/256 = 6 elems each
    __shared__ float red[256];
    float vals[HID / 256];
    float acc = 0.f;
#pragma unroll
    for (int i = 0; i < HID / 256; ++i) {
        const int idx = t + i * 256;
        const float v = x[b * HID + idx] + r[b * HID + idx];
        vals[i] = v;
        res_out[b * HID + idx] = v;
        acc += v * v;
    }
    red[t] = acc;
    __syncthreads();
    for (int s = 128; s > 0; s >>= 1) {
        if (t < s) red[t] += red[t + s];
        __syncthreads();
    }
    const float rstd = rsqrtf(red[0] / (float)HID + EPS_);
#pragma unroll
    for (int i = 0; i < HID / 256; ++i) {
        const int idx = t + i * 256;
        h_out[b * HID + idx] = vals[i] * rstd * w[idx];
    }
}

// ---------------- fp32 WMMA GEMM with async LDS + register pipelining -----
// C[64,N] = A[64,K] @ B[K,N] (+bias). One wave per 16-wide N strip.
//  - B strip: HBM -> LDS via GLOBAL_LOAD_ASYNC_TO_LDS_B128, double buffered,
//    with persistent per-chunk address registers (stride-advance, no rederive)
//  - A frags: register double buffered (loaded one tile ahead)
//  - B frags: pipelined one k-step ahead so ds latency hides under WMMAs
__device__ __forceinline__ void load_a_frags(const float* __restrict__ A, int K,
                                             int mrow, int khalf, int k0,
                                             v2f (&a)[8][4]) {
#pragma unroll
    for (int kk = 0; kk < 8; ++kk) {
#pragma unroll
        for (int m = 0; m < 4; ++m) {
            a[kk][m] = *(const v2f*)(A + (size_t)(mrow + 16 * m) * K + k0 + kk * 4 + khalf);
        }
    }
}

__device__ __forceinline__ void compute_tile(const float* __restrict__ bt,
                                             int khalf, int ncol,
                                             const v2f (&a)[8][4],
                                             v8f& acc0, v8f& acc1, v8f& acc2, v8f& acc3) {
    v2f bcur;
    bcur.x = bt[(khalf) * 16 + ncol];
    bcur.y = bt[(khalf + 1) * 16 + ncol];
#pragma unroll
    for (int kk = 0; kk < 8; ++kk) {
        v2f bnext;
        if (kk < 7) {   // prefetch next B fragment before the WMMA group
            bnext.x = bt[((kk + 1) * 4 + khalf) * 16 + ncol];
            bnext.y = bt[((kk + 1) * 4 + khalf + 1) * 16 + ncol];
        }
        acc0 = __builtin_amdgcn_wmma_f32_16x16x4_f32(false, a[kk][0], false, bcur, (short)0, acc0, false, false);
        acc1 = __builtin_amdgcn_wmma_f32_16x16x4_f32(false, a[kk][1], false, bcur, (short)0, acc1, false, false);
        acc2 = __builtin_amdgcn_wmma_f32_16x16x4_f32(false, a[kk][2], false, bcur, (short)0, acc2, false, false);
        acc3 = __builtin_amdgcn_wmma_f32_16x16x4_f32(false, a[kk][3], false, bcur, (short)0, acc3, false, false);
        if (kk < 7) bcur = bnext;
    }
}

__global__ void gemm_f32_wmma_async_kernel(const float* __restrict__ A,
                                           const float* __restrict__ B,
                                           const float* __restrict__ bias,
                                           float* __restrict__ C,
                                           int K, int N) {
    const int lane  = threadIdx.x;        // 0..31 (wave32)
    const int n0    = blockIdx.x * 16;
    const int mrow  = lane & 15;          // A: lane = M row
    const int khalf = (lane >> 4) * 2;    // K split across lane halves
    const int ncol  = lane & 15;          // B/D: lane = N col

    __shared__ float btile[2][TK * 16];   // 2 x 2KB B tiles

    const int ntiles = K / TK;            // even for all calls here

    // persistent per-chunk async-copy addresses (advance by one tile / issue)
    unsigned long long ga[4];
    unsigned           la[2][4];
    {
        const unsigned lbase = (unsigned)(uintptr_t)(&btile[0][0]);
#pragma unroll
        for (int c = 0; c < 4; ++c) {
            const int id   = lane + 32 * c;   // 128 x 16B chunks per tile
            const int row  = id >> 2;
            const int quad = id & 3;
            ga[c] = (unsigned long long)(uintptr_t)(B + (size_t)row * N + n0 + quad * 4);
            la[0][c] = lbase + (unsigned)(id * 16);
            la[1][c] = la[0][c] + (unsigned)(TK * 16 * 4);
        }
    }
    const unsigned long long gstep = (unsigned long long)TK * N * sizeof(float);

    // issue the next tile in sequence into LDS buffer `buf` (compile-time 0/1)
    auto issue = [&](int buf) {
#pragma unroll
        for (int c = 0; c < 4; ++c) {
            asm volatile("global_load_async_to_lds_b128 %0, %1, off"
                         :: "v"(la[buf][c]), "v"(ga[c]) : "memory");
            ga[c] += gstep;
        }
    };

    issue(0);
    if (ntiles > 1) issue(1);

    v2f aEven[8][4], aOdd[8][4];
    load_a_frags(A, K, mrow, khalf, 0, aEven);

    v8f acc0 = {}, acc1 = {}, acc2 = {}, acc3 = {};

    for (int t = 0; t < ntiles; t += 2) {
        // ---- even tile t : consumes aEven + btile[0], prefetches aOdd ----
        if (t + 1 < ntiles)
            asm volatile("s_wait_asynccnt 0x4" ::: "memory");
        else
            asm volatile("s_wait_asynccnt 0x0" ::: "memory");
        if (t + 1 < ntiles)
            load_a_frags(A, K, mrow, khalf, (t + 1) * TK, aOdd);
        compute_tile(&btile[0][0], khalf, ncol, aEven, acc0, acc1, acc2, acc3);
        if (t + 2 < ntiles) issue(0);

        // ---- odd tile t+1 : consumes aOdd + btile[1], prefetches aEven ---
        if (t + 1 < ntiles) {
            if (t + 2 < ntiles)
                asm volatile("s_wait_asynccnt 0x4" ::: "memory");
            else
                asm volatile("s_wait_asynccnt 0x0" ::: "memory");
            if (t + 2 < ntiles)
                load_a_frags(A, K, mrow, khalf, (t + 2) * TK, aEven);
            compute_tile(&btile[1][0], khalf, ncol, aOdd, acc0, acc1, acc2, acc3);
            if (t + 3 < ntiles) issue(1);
        }
    }

    const float bv = bias ? bias[n0 + ncol] : 0.f;
    const int mbase = (lane >> 4) * 8;    // D: lanes 16-31 hold rows M=8..15
#pragma unroll
    for (int i = 0; i < 8; ++i) {
        C[(size_t)( 0 + mbase + i) * N + n0 + ncol] = acc0[i] + bv;
        C[(size_t)(16 + mbase + i) * N + n0 + ncol] = acc1[i] + bv;
        C[(size_t)(32 + mbase + i) * N + n0 + ncol] = acc2[i] + bv;
        C[(size_t)(48 + mbase + i) * N + n0 + ncol] = acc3[i] + bv;
    }
}

// ---------------- NEOX RoPE on q and k (reads qkv, writes qr/kr) ----------
__global__ void rope_kernel(const int* __restrict__ positions,
                            const float* __restrict__ qkv,
                            float* __restrict__ qr,
                            float* __restrict__ kr) {
    const int b = blockIdx.x;
    const int t = threadIdx.x;            // 256
    const float pos = (float)positions[b];
    for (int w = t; w < (NH + NKV) * (HD / 2); w += 256) {
        const int head = w >> 6;          // 0..13
        const int i    = w & 63;          // pair index
        const float freq = __powf(1.0e6f, -(float)(2 * i) / (float)HD);
        const float ang  = pos * freq;
        const float c = __cosf(ang);
        const float s = __sinf(ang);
        const float* src;
        float* dst;
        if (head < NH) {
            src = qkv + (size_t)b * QKV_N + head * HD;
            dst = qr + ((size_t)b * NH + head) * HD;
        } else {
            src = qkv + (size_t)b * QKV_N + Q_SZ + (head - NH) * HD;
            dst = kr + ((size_t)b * NKV + (head - NH)) * HD;
        }
        const float x1 = src[i];
        const float x2 = src[i + HD / 2];
        dst[i]          = x1 * c - x2 * s;
        dst[i + HD / 2] = x2 * c + x1 * s;
    }
}

// ---------------- paged GQA decode attention -------------------------------
__global__ void attn_decode_kernel(const float* __restrict__ qr,
                                   const float* __restrict__ kr,
                                   const float* __restrict__ qkv,
                                   const float* __restrict__ kv_cache,
                                   const int* __restrict__ block_tables,
                                   const int* __restrict__ seq_lens,
                                   float* __restrict__ out) {
    const int h   = blockIdx.x;           // 0..11
    const int b   = blockIdx.y;           // 0..63
    const int kvh = h / GQA_G;
    const int t   = threadIdx.x;          // 256
    const int sl  = seq_lens[b];
    const int pos = sl - 1;
    const int* bt = block_tables + (size_t)b * MAXB;
    const size_t VOFF = (size_t)NBLK * BS_ * NKV * HD;   // v-plane offset

    __shared__ float sc[LMAX];
    __shared__ float red[256];
    __shared__ float sq[HD];

    if (t < HD) sq[t] = qr[((size_t)b * NH + h) * HD + t];
    __syncthreads();

    // phase 1: scores (new token's k comes from fresh projection: l == pos)
    float lmax = -3.0e38f;
    for (int l = t; l < sl; l += 256) {
        const float* kp;
        if (l == pos) {
            kp = kr + ((size_t)b * NKV + kvh) * HD;
        } else {
            const int blk = bt[l >> 4];
            kp = kv_cache + ((size_t)(blk * BS_ + (l & 15)) * NKV + kvh) * HD;
        }
        float s = 0.f;
#pragma unroll
        for (int d = 0; d < HD; d += 4) {
            const float4 k4 = *(const float4*)(kp + d);
            s += sq[d] * k4.x + sq[d + 1] * k4.y + sq[d + 2] * k4.z + sq[d + 3] * k4.w;
        }
        s *= ATT_SCALE;
        sc[l] = s;
        lmax = fmaxf(lmax, s);
    }
    red[t] = lmax;
    __syncthreads();
    for (int s = 128; s > 0; s >>= 1) {
        if (t < s) red[t] = fmaxf(red[t], red[t + s]);
        __syncthreads();
    }
    const float mx = red[0];
    __syncthreads();

    float lsum = 0.f;
    for (int l = t; l < sl; l += 256) {
        const float e = __expf(sc[l] - mx);
        sc[l] = e;
        lsum += e;
    }
    red[t] = lsum;
    __syncthreads();
    for (int s = 128; s > 0; s >>= 1) {
        if (t < s) red[t] += red[t + s];
        __syncthreads();
    }
    const float inv = 1.f / red[0];
    __syncthreads();

    // phase 2: probs @ V
    const int d    = t & (HD - 1);
    const int part = t >> 7;
    float acc = 0.f;
    for (int l = part; l < sl; l += 2) {
        const float* vp;
        if (l == pos) {
            vp = qkv + (size_t)b * QKV_N + Q_SZ + KV_SZ + kvh * HD;
        } else {
            const int blk = bt[l >> 4];
            vp = kv_cache + VOFF + ((size_t)(blk * BS_ + (l & 15)) * NKV + kvh) * HD;
        }
        acc += sc[l] * vp[d];
    }
    red[t] = acc;
    __syncthreads();
    if (part == 0)
        out[((size_t)b * NH + h) * HD + d] = (red[d] + red[d + HD]) * inv;
}

// ---------------- SwiGLU activation ---------------------------------------
__global__ void silu_mul_kernel(const float* __restrict__ gu,
                                float* __restrict__ act) {
    const int i = blockIdx.x * blockDim.x + threadIdx.x;
    if (i >= B_ * INTER) return;
    const int row = i / INTER;
    const int col = i - row * INTER;
    const float g = gu[(size_t)row * (2 * INTER) + col];
    const float u = gu[(size_t)row * (2 * INTER) + INTER + col];
    act[i] = (g / (1.f + __expf(-g))) * u;
}

// ---------------- launch ---------------------------------------------------
extern "C" void kernel_launch(void* const* d_in, const int* in_sizes, int n_in,
                              void* d_out, int out_size, void* d_ws, size_t ws_size,
                              hipStream_t stream) {
    const int*   positions    = (const int*)d_in[0];
    const float* hidden       = (const float*)d_in[1];
    const float* residual     = (const float*)d_in[2];
    const float* kv_cache     = (const float*)d_in[3];
    const int*   block_tables = (const int*)d_in[4];
    // d_in[5] slot_mapping: cache write folded into attention (l == pos)
    const int*   seq_lens     = (const int*)d_in[6];
    // d_in[7] is_prefill: statically 0 (decode)
    const float* w_qkv        = (const float*)d_in[8];
    const float* b_qkv        = (const float*)d_in[9];
    const float* w_o          = (const float*)d_in[10];
    const float* ln1_w        = (const float*)d_in[11];
    const float* ln2_w        = (const float*)d_in[12];
    const float* w_gate_up    = (const float*)d_in[13];
    const float* w_down       = (const float*)d_in[14];

    float* out = (float*)d_out;                 // [0:98304) h_mlp, [98304:196608) res2
    float* ws  = (float*)d_ws;

    float* res   = ws;                          // 64*1536
    float* h1    = ws + 98304;                  // 64*1536
    float* qkv   = ws + 196608;                 // 64*2048
    float* qr    = ws + 327680;                 // 64*1536
    float* kr    = ws + 425984;                 // 64*256
    float* attn  = ws + 442368;                 // 64*1536
    float* o_out = ws + 540672;                 // 64*1536
    float* h2    = ws + 638976;                 // 64*1536
    float* gu    = ws + 737280;                 // 64*17920
    float* act   = ws + 1884160;                // 64*8960

    // 1. res = hs + residual ; h1 = rms(res) * ln1_w
    add_rmsnorm_kernel<<<B_, 256, 0, stream>>>(hidden, residual, ln1_w, res, h1);

    // 2. qkv = h1 @ w_qkv + b_qkv     [64,1536]x[1536,2048]
    gemm_f32_wmma_async_kernel<<<QKV_N / 16, 32, 0, stream>>>(h1, w_qkv, b_qkv, qkv, HID, QKV_N);

    // 3. NEOX RoPE
    rope_kernel<<<B_, 256, 0, stream>>>(positions, qkv, qr, kr);

    // 4. paged GQA decode attention
    dim3 ag(NH, B_);
    attn_decode_kernel<<<ag, 256, 0, stream>>>(qr, kr, qkv, kv_cache, block_tables,
                                               seq_lens, attn);

    // 5. o_out = attn @ w_o           [64,1536]x[1536,1536]
    gemm_f32_wmma_async_kernel<<<HID / 16, 32, 0, stream>>>(attn, w_o, nullptr, o_out, Q_SZ, HID);

    // 6. res2 = o_out + res (-> d_out upper half) ; h2 = rms(res2) * ln2_w
    add_rmsnorm_kernel<<<B_, 256, 0, stream>>>(o_out, res, ln2_w, out + B_ * HID, h2);

    // 7. gu = h2 @ w_gate_up          [64,1536]x[1536,17920]
    gemm_f32_wmma_async_kernel<<<(2 * INTER) / 16, 32, 0, stream>>>(h2, w_gate_up, nullptr, gu,
                                                                    HID, 2 * INTER);

    // 8. act = silu(gate) * up
    silu_mul_kernel<<<(B_ * INTER + 255) / 256, 256, 0, stream>>>(gu, act);

    // 9. h_mlp = act @ w_down (-> d_out lower half)   [64,8960]x[8960,1536]
    gemm_f32_wmma_async_kernel<<<HID / 16, 32, 0, stream>>>(act, w_down, nullptr, out, INTER, HID);
}